// SpatialRefiner_17128329576753
// MI455X (gfx1250) — compile-verified
//
#include <hip/hip_runtime.h>
#include <hip/hip_bf16.h>
#include <math.h>

#define Bsz   32
#define Himg  512
#define Wimg  512
#define HWpx  (Himg * Wimg)      // 262144
#define THETA_SPATIAL 20.0f
#define NBINS_MAX 2048

// workspace layout (bytes)
#define WS_H_OFF      0          // 32*9 floats   : per-batch homography H (row-major)
#define WS_PREFIX_OFF 1152       // 32 uints      : radix-select prefix
#define WS_K_OFF      1280       // 32 uints      : remaining rank
#define WS_MED_OFF    1408       // 32 floats     : final median
#define WS_HIST_OFF   1536       // 32*2048 uints : per-batch histograms

typedef __attribute__((ext_vector_type(2))) float v2f;
typedef __attribute__((ext_vector_type(8))) float v8f;

// ---------------------------------------------------------------------------
// Element (r, i) of the 8x9 DLT matrix A for one batch.
// rows interleaved: r even -> r1 of point r/2, r odd -> r2 of point r/2.
// ---------------------------------------------------------------------------
__device__ __forceinline__ float dlt_elem(int r, int i,
                                          const float* __restrict__ s,
                                          const float* __restrict__ d) {
  int p = r >> 1;
  float x = s[p * 2 + 0], y = s[p * 2 + 1];
  float u = d[p * 2 + 0], v = d[p * 2 + 1];
  if (i > 8) return 0.0f;
  if ((r & 1) == 0) {
    switch (i) {
      case 0: return -x;   case 1: return -y;   case 2: return -1.0f;
      case 6: return u*x;  case 7: return u*y;  case 8: return u;
      default: return 0.0f;
    }
  } else {
    switch (i) {
      case 3: return -x;   case 4: return -y;   case 5: return -1.0f;
      case 6: return v*x;  case 7: return v*y;  case 8: return v;
      default: return 0.0f;
    }
  }
}

// ---------------------------------------------------------------------------
// One wave (32 lanes) per batch:
//   M = A^T A via two V_WMMA_F32_16X16X4_F32 k-steps (16x16x8, 9x9 live),
//   then lane 0 runs a cyclic Jacobi eigensolver on the 9x9 Gram matrix and
//   writes H = nullvec / nullvec[8] to workspace.
// f32 A-operand layout (16x4): lane<16 holds K=0,1; lane>=16 holds K=2,3.
// Mirrored B-operand layout (4x16) makes the per-lane data of A^T (as A-op)
// and A (as B-op) identical -> a single float2 feeds both operands.
// ---------------------------------------------------------------------------
__global__ __launch_bounds__(32) void homography_kernel(
    const float* __restrict__ src, const float* __restrict__ dst,
    float* __restrict__ ws_f) {
  const int b    = blockIdx.x;
  const int lane = threadIdx.x;
  const int i    = lane & 15;   // M index of A-op == N index of B-op
  const int hi   = lane >> 4;

  const float* s = src + b * 8;
  const float* d = dst + b * 8;

  __shared__ float Msh[81];

  v8f acc = {0.f, 0.f, 0.f, 0.f, 0.f, 0.f, 0.f, 0.f};

  v2f op0;  // k-step 0: DLT rows 0..3
  op0.x = dlt_elem(0 + 2 * hi + 0, i, s, d);
  op0.y = dlt_elem(0 + 2 * hi + 1, i, s, d);
  acc = __builtin_amdgcn_wmma_f32_16x16x4_f32(false, op0, false, op0,
                                              (short)0, acc, false, false);
  v2f op1;  // k-step 1: DLT rows 4..7
  op1.x = dlt_elem(4 + 2 * hi + 0, i, s, d);
  op1.y = dlt_elem(4 + 2 * hi + 1, i, s, d);
  acc = __builtin_amdgcn_wmma_f32_16x16x4_f32(false, op1, false, op1,
                                              (short)0, acc, false, false);

  // D layout: VGPR r, lanes 0-15 -> M=r, lanes 16-31 -> M=r+8; N = lane&15.
#pragma unroll
  for (int r = 0; r < 8; ++r) {
    int row = r + 8 * hi;
    if (row < 9 && i < 9) Msh[row * 9 + i] = acc[r];
  }
  __syncthreads();

  if (lane == 0) {
    float M[81], V[81];
    for (int k = 0; k < 81; ++k) { M[k] = Msh[k]; V[k] = 0.0f; }
    for (int k = 0; k < 9; ++k) V[k * 9 + k] = 1.0f;

    for (int sweep = 0; sweep < 15; ++sweep) {
      for (int p = 0; p < 8; ++p) {
        for (int q = p + 1; q < 9; ++q) {
          float apq = M[p * 9 + q];
          float app = M[p * 9 + p], aqq = M[q * 9 + q];
          if (fabsf(apq) <= 1e-12f * (fabsf(app) + fabsf(aqq)) + 1e-30f) continue;
          float tau = (aqq - app) / (2.0f * apq);
          float t   = copysignf(1.0f, tau) / (fabsf(tau) + sqrtf(1.0f + tau * tau));
          float c   = 1.0f / sqrtf(1.0f + t * t);
          float sn  = t * c;
          for (int k = 0; k < 9; ++k) {       // M <- M G (columns p,q)
            float mkp = M[k * 9 + p], mkq = M[k * 9 + q];
            M[k * 9 + p] = c * mkp - sn * mkq;
            M[k * 9 + q] = sn * mkp + c * mkq;
          }
          for (int k = 0; k < 9; ++k) {       // M <- G^T M (rows p,q)
            float mpk = M[p * 9 + k], mqk = M[q * 9 + k];
            M[p * 9 + k] = c * mpk - sn * mqk;
            M[q * 9 + k] = sn * mpk + c * mqk;
          }
          for (int k = 0; k < 9; ++k) {       // V <- V G
            float vkp = V[k * 9 + p], vkq = V[k * 9 + q];
            V[k * 9 + p] = c * vkp - sn * vkq;
            V[k * 9 + q] = sn * vkp + c * vkq;
          }
        }
      }
    }
    int jm = 0; float mn = M[0];
    for (int k = 1; k < 9; ++k) {
      float dk = M[k * 9 + k];
      if (dk < mn) { mn = dk; jm = k; }
    }
    float inv = 1.0f / V[8 * 9 + jm];
    float* Hout = ws_f + (size_t)WS_H_OFF / 4 + b * 9;
    for (int k = 0; k < 9; ++k) Hout[k] = V[k * 9 + jm] * inv;
  }
}

// ---------------------------------------------------------------------------
__global__ void init_kernel(unsigned int* __restrict__ ws_u) {
  int idx = blockIdx.x * blockDim.x + threadIdx.x;
  if (idx < Bsz * NBINS_MAX) ws_u[WS_HIST_OFF / 4 + idx] = 0u;
  if (idx < Bsz) {
    ws_u[WS_PREFIX_OFF / 4 + idx] = 0u;
    ws_u[WS_K_OFF / 4 + idx]      = (HWpx - 1) / 2 + 1;  // 1-indexed rank 131072
  }
}

// ---------------------------------------------------------------------------
// diff for 4 consecutive pixels (float4 path), shared by hist & apply.
// ---------------------------------------------------------------------------
__device__ __forceinline__ void pixel_quad(
    const float* __restrict__ of, const float* __restrict__ Hb,
    int b, int n0, float* __restrict__ ofx, float* __restrict__ ofy,
    float* __restrict__ fx, float* __restrict__ fy,
    float* __restrict__ diff) {
  float h00 = Hb[0], h01 = Hb[1], h02 = Hb[2];
  float h10 = Hb[3], h11 = Hb[4], h12 = Hb[5];
  float h20 = Hb[6], h21 = Hb[7], h22 = Hb[8];

  float4 f0 = *(const float4*)(of + (size_t)b * 2 * HWpx + n0);
  float4 f1 = *(const float4*)(of + (size_t)b * 2 * HWpx + HWpx + n0);
  ofx[0] = f0.x; ofx[1] = f0.y; ofx[2] = f0.z; ofx[3] = f0.w;
  ofy[0] = f1.x; ofy[1] = f1.y; ofy[2] = f1.z; ofy[3] = f1.w;

  float yf = (float)(n0 >> 9);  // W=512; quad never crosses a row
#pragma unroll
  for (int j = 0; j < 4; ++j) {
    float xf = (float)((n0 + j) & 511);
    float t1 = fmaf(h00, xf, fmaf(h01, yf, h02));
    float t2 = fmaf(h10, xf, fmaf(h11, yf, h12));
    float t3 = fmaf(h20, xf, fmaf(h21, yf, h22));
    float inv = 1.0f / (t3 + 1e-8f);
    fx[j] = t1 * inv - xf;
    fy[j] = t2 * inv - yf;
    float dx = ofx[j] - fx[j], dy = ofy[j] - fy[j];
    diff[j] = sqrtf(dx * dx + dy * dy);
  }
}

// ---------------------------------------------------------------------------
// One radix-select round: LDS-privatized per-batch histogram of key digits.
// ---------------------------------------------------------------------------
__global__ __launch_bounds__(256) void hist_kernel(
    const float* __restrict__ of, const float* __restrict__ ws_f,
    unsigned int* __restrict__ ws_u, int shift, int nbins, int round) {
  int b  = blockIdx.y;
  int t  = blockIdx.x * 256 + threadIdx.x;
  int n0 = t * 4;

  __shared__ unsigned int lh[NBINS_MAX];
  for (int k = threadIdx.x; k < nbins; k += 256) lh[k] = 0u;
  __syncthreads();

  const float* Hb = ws_f + (size_t)WS_H_OFF / 4 + b * 9;
  unsigned int prefix = ws_u[WS_PREFIX_OFF / 4 + b];

  float ofx[4], ofy[4], fx[4], fy[4], diff[4];
  pixel_quad(of, Hb, b, n0, ofx, ofy, fx, fy, diff);

#pragma unroll
  for (int j = 0; j < 4; ++j) {
    unsigned int key = __float_as_uint(diff[j]);  // diff>=0 -> bits monotone
    bool ok = (round == 0) ? true
            : (round == 1) ? ((key >> 21) == (prefix >> 21))
                           : ((key >> 10) == (prefix >> 10));
    if (ok) atomicAdd(&lh[(key >> shift) & (unsigned)(nbins - 1)], 1u);
  }
  __syncthreads();

  unsigned int* gh = ws_u + WS_HIST_OFF / 4 + (size_t)b * NBINS_MAX;
  for (int k = threadIdx.x; k < nbins; k += 256) {
    unsigned int v = lh[k];
    if (v) atomicAdd(&gh[k], v);
  }
}

// ---------------------------------------------------------------------------
// 32 threads: per-batch scan of the histogram -> select digit, update state,
// zero histogram for the next round.
// ---------------------------------------------------------------------------
__global__ __launch_bounds__(32) void scan_kernel(
    unsigned int* __restrict__ ws_u, float* __restrict__ ws_f,
    int shift, int nbins, int last) {
  int b = threadIdx.x;
  if (b >= Bsz) return;
  unsigned int* gh = ws_u + WS_HIST_OFF / 4 + (size_t)b * NBINS_MAX;
  unsigned int k   = ws_u[WS_K_OFF / 4 + b];
  unsigned int cum = 0;
  unsigned int bin = (unsigned)(nbins - 1);
  for (int i = 0; i < nbins; ++i) {
    unsigned int c = gh[i];
    if (cum + c >= k) { bin = (unsigned)i; break; }
    cum += c;
  }
  for (int i = 0; i < nbins; ++i) gh[i] = 0u;
  unsigned int prefix = ws_u[WS_PREFIX_OFF / 4 + b] | (bin << shift);
  ws_u[WS_PREFIX_OFF / 4 + b] = prefix;
  ws_u[WS_K_OFF / 4 + b]      = k - cum;
  if (last) ws_f[WS_MED_OFF / 4 + b] = __uint_as_float(prefix);
}

// ---------------------------------------------------------------------------
// Fused final pass: recompute hflow/diff, threshold at median+20, write
// refined flow (float4) and mask-as-float.
// ---------------------------------------------------------------------------
__global__ __launch_bounds__(256) void apply_kernel(
    const float* __restrict__ of, const float* __restrict__ ws_f,
    float* __restrict__ out) {
  int b  = blockIdx.y;
  int t  = blockIdx.x * 256 + threadIdx.x;
  int n0 = t * 4;

  const float* Hb = ws_f + (size_t)WS_H_OFF / 4 + b * 9;
  float thresh = ws_f[WS_MED_OFF / 4 + b] + THETA_SPATIAL;

  float ofx[4], ofy[4], fx[4], fy[4], diff[4];
  pixel_quad(of, Hb, b, n0, ofx, ofy, fx, fy, diff);

  float r0[4], r1[4], mk[4];
#pragma unroll
  for (int j = 0; j < 4; ++j) {
    bool outl = diff[j] > thresh;
    r0[j] = outl ? fx[j] : ofx[j];
    r1[j] = outl ? fy[j] : ofy[j];
    mk[j] = outl ? 1.0f : 0.0f;
  }
  *(float4*)(out + (size_t)b * 2 * HWpx + n0)        = make_float4(r0[0], r0[1], r0[2], r0[3]);
  *(float4*)(out + (size_t)b * 2 * HWpx + HWpx + n0) = make_float4(r1[0], r1[1], r1[2], r1[3]);
  *(float4*)(out + (size_t)Bsz * 2 * HWpx + (size_t)b * HWpx + n0)
      = make_float4(mk[0], mk[1], mk[2], mk[3]);
}

// ---------------------------------------------------------------------------
extern "C" void kernel_launch(void* const* d_in, const int* in_sizes, int n_in,
                              void* d_out, int out_size, void* d_ws, size_t ws_size,
                              hipStream_t stream) {
  const float* of  = (const float*)d_in[0];  // (B,2,H,W)
  const float* src = (const float*)d_in[1];  // (B,4,2)
  const float* dst = (const float*)d_in[2];  // (B,4,2)
  float* out = (float*)d_out;                // refined (B,2,H,W) ++ mask (B,1,H,W)
  float*        ws_f = (float*)d_ws;
  unsigned int* ws_u = (unsigned int*)d_ws;

  homography_kernel<<<Bsz, 32, 0, stream>>>(src, dst, ws_f);

  int initN = Bsz * NBINS_MAX;
  init_kernel<<<(initN + 255) / 256, 256, 0, stream>>>(ws_u);

  dim3 grid(HWpx / 4 / 256, Bsz);  // (256, 32)
  hist_kernel<<<grid, 256, 0, stream>>>(of, ws_f, ws_u, 21, 2048, 0);
  scan_kernel<<<1, 32, 0, stream>>>(ws_u, ws_f, 21, 2048, 0);
  hist_kernel<<<grid, 256, 0, stream>>>(of, ws_f, ws_u, 10, 2048, 1);
  scan_kernel<<<1, 32, 0, stream>>>(ws_u, ws_f, 10, 2048, 0);
  hist_kernel<<<grid, 256, 0, stream>>>(of, ws_f, ws_u, 0, 1024, 2);
  scan_kernel<<<1, 32, 0, stream>>>(ws_u, ws_f, 0, 1024, 1);

  apply_kernel<<<grid, 256, 0, stream>>>(of, ws_f, out);
}